// KipfNet_78039555768470
// MI455X (gfx1250) — compile-verified
//
#include <hip/hip_runtime.h>

typedef float v2f __attribute__((ext_vector_type(2)));
typedef float v8f __attribute__((ext_vector_type(8)));

// ---------------------------------------------------------------- utilities
__global__ __launch_bounds__(256) void zero_f32(float* __restrict__ p, int n4) {
  int i = blockIdx.x * 256 + threadIdx.x;
  if (i < n4) ((float4*)p)[i] = make_float4(0.f, 0.f, 0.f, 0.f);
}

// dst = -src   (accumulator init: folds the "- Tx0" of the Chebyshev recursion)
__global__ __launch_bounds__(256) void neg_copy(float* __restrict__ dst,
                                                const float* __restrict__ src, int n4) {
  int i = blockIdx.x * 256 + threadIdx.x;
  if (i < n4) {
    float4 v = ((const float4*)src)[i];
    ((float4*)dst)[i] = make_float4(-v.x, -v.y, -v.z, -v.w);
  }
}

__global__ __launch_bounds__(256) void deg_accum(const int* __restrict__ col,
                                                 float* __restrict__ deg, int E) {
  int e = blockIdx.x * 256 + threadIdx.x;
  if (e < E) atomicAdd(&deg[col[e]], 1.0f);
}

__global__ __launch_bounds__(256) void finish_dis(float* __restrict__ deg, int N) {
  int i = blockIdx.x * 256 + threadIdx.x;
  if (i < N) {
    float d = deg[i];
    deg[i] = (d > 0.f) ? rsqrtf(fmaxf(d, 1.0f)) : 0.f;
  }
}

__global__ __launch_bounds__(256) void edge_norm(const int* __restrict__ row,
                                                 const int* __restrict__ col,
                                                 const float* __restrict__ dis,
                                                 float* __restrict__ norm, int E) {
  int e = blockIdx.x * 256 + threadIdx.x;
  if (e < E) norm[e] = -(dis[row[e]] * dis[col[e]]);
}

// dst[col] += scale * norm[e] * src[row]   (6 threads per edge, float4 each)
__global__ __launch_bounds__(256) void scatter_prop(const float* __restrict__ src,
                                                    float* __restrict__ dst,
                                                    const int* __restrict__ row,
                                                    const int* __restrict__ col,
                                                    const float* __restrict__ norm,
                                                    float scale, int total) {
  int gid = blockIdx.x * 256 + threadIdx.x;
  if (gid >= total) return;
  int e  = gid / 6;
  int f4 = (gid - e * 6) * 4;
  int r = row[e], c = col[e];
  float w = norm[e] * scale;
  float4 v = *(const float4*)(src + r * 24 + f4);
  float* dp = dst + c * 24 + f4;
  atomicAdd(dp + 0, w * v.x);
  atomicAdd(dp + 1, w * v.y);
  atomicAdd(dp + 2, w * v.z);
  atomicAdd(dp + 3, w * v.w);
}

// --------------------------------------------------------------------------
// h[32 rows] = sum_k Tx_k @ W1[k]   -- single pass over all 6 Chebyshev bases
// One wave32 per 32 rows (two M-tiles), 4 N-tiles cover H=64. f32 WMMA.
// B fragments are shared by both M-tiles (register reuse).
// --------------------------------------------------------------------------
__global__ __launch_bounds__(32) void gemm_all(const float* __restrict__ t0,
                                               const float* __restrict__ t1,
                                               const float* __restrict__ t2,
                                               const float* __restrict__ t3,
                                               const float* __restrict__ t4,
                                               const float* __restrict__ t5,
                                               const float* __restrict__ W1,
                                               float* __restrict__ h, int N) {
  const float* Tx[6] = {t0, t1, t2, t3, t4, t5};
  int lane  = threadIdx.x;
  int cn    = lane & 15;
  int khalf = (lane >> 4) << 1;              // 0 or 2
  int m0    = blockIdx.x * 32;
  int mA0   = m0 + (lane & 15);
  int mA1   = mA0 + 16;
  if (mA0 > N - 1) mA0 = N - 1;              // keep EXEC full for WMMA
  if (mA1 > N - 1) mA1 = N - 1;

  v8f c[2][4] = {};
#pragma unroll
  for (int k = 0; k < 6; ++k) {
    const float* T  = Tx[k];
    const float* Wk = W1 + k * 24 * 64;
#pragma unroll
    for (int k0 = 0; k0 < 24; k0 += 4) {
      const float* ap0 = T + mA0 * 24 + k0 + khalf;
      const float* ap1 = T + mA1 * 24 + k0 + khalf;
      v2f a0; a0.x = ap0[0]; a0.y = ap0[1];
      v2f a1; a1.x = ap1[0]; a1.y = ap1[1];
#pragma unroll
      for (int t = 0; t < 4; ++t) {
        const float* bp = Wk + (k0 + khalf) * 64 + t * 16 + cn;
        v2f b; b.x = bp[0]; b.y = bp[64];
        c[0][t] = __builtin_amdgcn_wmma_f32_16x16x4_f32(false, a0, false, b,
                                                        (short)0, c[0][t], false, false);
        c[1][t] = __builtin_amdgcn_wmma_f32_16x16x4_f32(false, a1, false, b,
                                                        (short)0, c[1][t], false, false);
      }
    }
  }

  int cmb = m0 + ((lane >> 4) << 3);         // +0 or +8 for C/D rows
#pragma unroll
  for (int mt = 0; mt < 2; ++mt)
#pragma unroll
    for (int t = 0; t < 4; ++t)
      for (int j = 0; j < 8; ++j) {
        int r = cmb + mt * 16 + j;
        if (r < N) h[r * 64 + t * 16 + cn] = c[mt][t][j];
      }
}

// ------------------------------------------------- BN column stats (sum, sumsq)
__global__ __launch_bounds__(256) void bn_stats(const float* __restrict__ h,
                                                float* __restrict__ stats, int N) {
  int lane = threadIdx.x;
  int c    = lane & 63;
  int rg   = lane >> 6;                      // 0..3
  float s = 0.f, sq = 0.f;
  for (int r = blockIdx.x * 4 + rg; r < N; r += gridDim.x * 4) {
    float v = h[r * 64 + c];
    s += v; sq += v * v;
  }
  __shared__ float ls[256], lq[256];
  ls[lane] = s; lq[lane] = sq;
  __syncthreads();
  if (lane < 64) {
    s  = ls[lane] + ls[lane + 64] + ls[lane + 128] + ls[lane + 192];
    sq = lq[lane] + lq[lane + 64] + lq[lane + 128] + lq[lane + 192];
    atomicAdd(&stats[lane], s);
    atomicAdd(&stats[64 + lane], sq);
  }
}

// ----------------------------- y = relu(BN(h)) @ Wmix + bmix  (f32 WMMA, OUT padded)
__global__ __launch_bounds__(32) void bn_relu_mix(const float* __restrict__ h,
                                                  const float* __restrict__ stats,
                                                  const float* __restrict__ gamma,
                                                  const float* __restrict__ beta,
                                                  const float* __restrict__ Wmix,
                                                  const float* __restrict__ bmix,
                                                  float* __restrict__ y, int N) {
  __shared__ float s_scale[64], s_shift[64];
  int lane = threadIdx.x;
  float inv_n = 1.0f / (float)N;
  for (int c = lane; c < 64; c += 32) {      // uniform 2 iterations, EXEC stays full
    float mean = stats[c] * inv_n;
    float var  = stats[64 + c] * inv_n - mean * mean;
    float g    = gamma[c] * rsqrtf(var + 1e-5f);
    s_scale[c] = g;
    s_shift[c] = beta[c] - mean * g;
  }
  __syncthreads();

  int m0    = blockIdx.x * 16;
  int cn    = lane & 15;
  int khalf = (lane >> 4) << 1;
  int mrow  = m0 + (lane & 15);
  if (mrow > N - 1) mrow = N - 1;

  v8f c = {};
#pragma unroll
  for (int k0 = 0; k0 < 64; k0 += 4) {
    int kk = k0 + khalf;
    float a0 = fmaf(h[mrow * 64 + kk],     s_scale[kk],     s_shift[kk]);
    float a1 = fmaf(h[mrow * 64 + kk + 1], s_scale[kk + 1], s_shift[kk + 1]);
    v2f a; a.x = fmaxf(a0, 0.f); a.y = fmaxf(a1, 0.f);
    v2f b;
    b.x = (cn < 6) ? Wmix[kk * 6 + cn]       : 0.f;   // pad OUT 6 -> 16
    b.y = (cn < 6) ? Wmix[(kk + 1) * 6 + cn] : 0.f;
    c = __builtin_amdgcn_wmma_f32_16x16x4_f32(false, a, false, b,
                                              (short)0, c, false, false);
  }
  int cmb = m0 + ((lane >> 4) << 3);
  if (cn < 6) {                              // stores only, after all WMMAs
    float bb = bmix[cn];
    for (int j = 0; j < 8; ++j) {
      int r = cmb + j;
      if (r < N) y[r * 6 + cn] = c[j] + bb;
    }
  }
}

// ---------------------------------------------------------------- launcher
extern "C" void kernel_launch(void* const* d_in, const int* in_sizes, int n_in,
                              void* d_out, int out_size, void* d_ws, size_t ws_size,
                              hipStream_t stream) {
  const int N = in_sizes[0] / 24;
  const int E = in_sizes[1] / 2;

  const float* x     = (const float*)d_in[0];
  const int*   ei    = (const int*)d_in[1];
  const int*   row   = ei;
  const int*   col   = ei + E;
  const float* W1    = (const float*)d_in[2];   // [6][24][64]
  // d_in[3] = b1: provably cancelled by batch-stat BatchNorm (per-column shift)
  const float* gamma = (const float*)d_in[4];
  const float* beta  = (const float*)d_in[5];
  const float* Wmix  = (const float*)d_in[6];   // [64][6]
  const float* bmix  = (const float*)d_in[7];
  float*       y     = (float*)d_out;

  char* ws = (char*)d_ws;
  float* buf1  = (float*)ws;  ws += (size_t)N * 24 * 4;   // Tx1..Tx5 all kept live
  float* buf2  = (float*)ws;  ws += (size_t)N * 24 * 4;
  float* buf3  = (float*)ws;  ws += (size_t)N * 24 * 4;
  float* buf4  = (float*)ws;  ws += (size_t)N * 24 * 4;
  float* buf5  = (float*)ws;  ws += (size_t)N * 24 * 4;
  float* h     = (float*)ws;  ws += (size_t)N * 64 * 4;
  float* dis   = (float*)ws;  ws += (size_t)N * 4;        // deg -> dis in place
  float* norm  = (float*)ws;  ws += (size_t)E * 4;
  float* stats = (float*)ws;                               // 128 floats

  const int nTx4  = (N * 24) / 4;
  const int gTx   = (nTx4 + 255) / 256;
  const int gE    = (E + 255) / 256;
  const int gN    = (N + 255) / 256;
  const int gScat = (E * 6 + 255) / 256;

  // ---- graph normalization: deg -> dis -> per-edge norm
  zero_f32<<<(N / 4 + 255) / 256, 256, 0, stream>>>(dis, N / 4);
  deg_accum<<<gE, 256, 0, stream>>>(col, dis, E);
  finish_dis<<<gN, 256, 0, stream>>>(dis, N);
  edge_norm<<<gE, 256, 0, stream>>>(row, col, dis, norm, E);

  // ---- Chebyshev recursion: Tx_{k} = 2*prop(Tx_{k-1}) - Tx_{k-2}
  //      "- Tx_{k-2}" folded into accumulator init via neg_copy.
  // Tx1 = prop(x)
  zero_f32<<<gTx, 256, 0, stream>>>(buf1, nTx4);
  scatter_prop<<<gScat, 256, 0, stream>>>(x, buf1, row, col, norm, 1.0f, E * 6);
  // Tx2 = 2*prop(Tx1) - x
  neg_copy<<<gTx, 256, 0, stream>>>(buf2, x, nTx4);
  scatter_prop<<<gScat, 256, 0, stream>>>(buf1, buf2, row, col, norm, 2.0f, E * 6);
  // Tx3 = 2*prop(Tx2) - Tx1
  neg_copy<<<gTx, 256, 0, stream>>>(buf3, buf1, nTx4);
  scatter_prop<<<gScat, 256, 0, stream>>>(buf2, buf3, row, col, norm, 2.0f, E * 6);
  // Tx4 = 2*prop(Tx3) - Tx2
  neg_copy<<<gTx, 256, 0, stream>>>(buf4, buf2, nTx4);
  scatter_prop<<<gScat, 256, 0, stream>>>(buf3, buf4, row, col, norm, 2.0f, E * 6);
  // Tx5 = 2*prop(Tx4) - Tx3
  neg_copy<<<gTx, 256, 0, stream>>>(buf5, buf3, nTx4);
  scatter_prop<<<gScat, 256, 0, stream>>>(buf4, buf5, row, col, norm, 2.0f, E * 6);

  // ---- h = sum_k Tx_k @ W1[k]   (single fused WMMA pass, writes h once)
  gemm_all<<<(N + 31) / 32, 32, 0, stream>>>(x, buf1, buf2, buf3, buf4, buf5,
                                             W1, h, N);

  // ---- BatchNorm stats + fused BN->ReLU->mix
  zero_f32<<<1, 256, 0, stream>>>(stats, 32);
  bn_stats<<<1024, 256, 0, stream>>>(h, stats, N);
  bn_relu_mix<<<(N + 15) / 16, 32, 0, stream>>>(h, stats, gamma, beta,
                                                Wmix, bmix, y, N);
}